// GatedAttention_75814762709421
// MI455X (gfx1250) — compile-verified
//
#include <hip/hip_runtime.h>
#include <hip/hip_bf16.h>

typedef __attribute__((ext_vector_type(2))) float v2f;
typedef __attribute__((ext_vector_type(8))) float v8f;

#define BATCH   8
#define NTOK    16384
#define DIM     512
#define CHUNKS  64                 // workgroups per batch
#define TOKWG   (NTOK / CHUNKS)    // 256 tokens per workgroup
#define THREADS 256
#define WAVES   (THREADS / 32)     // 8 wave32
#define TOKWAVE (TOKWG / WAVES)    // 32 tokens per wave

// Full-wave (32 lane) allreduce-add using DS_SWIZZLE_B32 group-of-32 xor mode.
// offset = xor_mask<<10 | or_mask<<5 | and_mask ; and_mask=0x1f keeps all lanes.
__device__ __forceinline__ float wave_allreduce_add(float v) {
    v += __int_as_float(__builtin_amdgcn_ds_swizzle(__float_as_int(v), 0x401F)); // xor 16
    v += __int_as_float(__builtin_amdgcn_ds_swizzle(__float_as_int(v), 0x201F)); // xor 8
    v += __int_as_float(__builtin_amdgcn_ds_swizzle(__float_as_int(v), 0x101F)); // xor 4
    v += __int_as_float(__builtin_amdgcn_ds_swizzle(__float_as_int(v), 0x081F)); // xor 2
    v += __int_as_float(__builtin_amdgcn_ds_swizzle(__float_as_int(v), 0x041F)); // xor 1
    return v;
}

__device__ __forceinline__ float dot4(float4 a, float4 b) {
    float r = a.x * b.x;
    r = fmaf(a.y, b.y, r);
    r = fmaf(a.z, b.z, r);
    r = fmaf(a.w, b.w, r);
    return r;
}

__global__ void zero_ws_kernel(float* __restrict__ ws, int n) {
    int i = blockIdx.x * blockDim.x + threadIdx.x;
    if (i < n) ws[i] = 0.0f;
}

__global__ void __launch_bounds__(THREADS)
gated_attn_main(const float* __restrict__ x,
                const float* __restrict__ Wa, const float* __restrict__ ba,
                const float* __restrict__ Wg, const float* __restrict__ bg,
                float* __restrict__ ctx_acc /* [BATCH][DIM] */,
                float* __restrict__ l_acc   /* [BATCH] */) {
    __shared__ float w_s[TOKWG];
    __shared__ float lpart[WAVES];

    const int tid  = threadIdx.x;
    const int lane = tid & 31;
    const int wave = tid >> 5;
    const int b     = blockIdx.x / CHUNKS;
    const int chunk = blockIdx.x % CHUNKS;
    const int tok0  = chunk * TOKWG;

    const float* xb = x + (size_t)b * NTOK * DIM;

    // ---------------- Phase 1: scores -> weights (single HBM pass over x) ----
    // Per-lane 16-float slice of the projection vectors.
    const int d0 = lane * 16;
    float4 wa0 = *(const float4*)(Wa + d0 + 0);
    float4 wa1 = *(const float4*)(Wa + d0 + 4);
    float4 wa2 = *(const float4*)(Wa + d0 + 8);
    float4 wa3 = *(const float4*)(Wa + d0 + 12);
    float4 wg0 = *(const float4*)(Wg + d0 + 0);
    float4 wg1 = *(const float4*)(Wg + d0 + 4);
    float4 wg2 = *(const float4*)(Wg + d0 + 8);
    float4 wg3 = *(const float4*)(Wg + d0 + 12);
    const float biasA = ba[0];
    const float biasG = bg[0];

    float lsum = 0.0f;
    for (int i = 0; i < TOKWAVE; ++i) {
        const int tl = wave * TOKWAVE + i;                 // local token id
        const float* row = xb + (size_t)(tok0 + tl) * DIM + d0;
        float4 x0 = *(const float4*)(row + 0);
        float4 x1 = *(const float4*)(row + 4);
        float4 x2 = *(const float4*)(row + 8);
        float4 x3 = *(const float4*)(row + 12);
        if (i + 1 < TOKWAVE) __builtin_prefetch(row + DIM, 0, 1);

        float da = dot4(x0, wa0) + dot4(x1, wa1) + dot4(x2, wa2) + dot4(x3, wa3);
        float dg = dot4(x0, wg0) + dot4(x1, wg1) + dot4(x2, wg2) + dot4(x3, wg3);
        da = wave_allreduce_add(da) + biasA;
        dg = wave_allreduce_add(dg) + biasG;

        // score = tanh(da) * sigmoid(dg) in (-1,1) -> exp() safe without max shift
        float ea = __expf(2.0f * da);
        float th = (ea - 1.0f) / (ea + 1.0f);
        float sg = 1.0f / (1.0f + __expf(-dg));
        float w  = __expf(th * sg);                        // unnormalized weight

        if (lane == 0) w_s[tl] = w;                        // wave-uniform value
        lsum += w;
    }
    if (lane == 0) lpart[wave] = lsum;
    __syncthreads();
    if (tid == 0) {
        float s = 0.0f;
        for (int i = 0; i < WAVES; ++i) s += lpart[i];
        atomicAdd(&l_acc[b], s);
    }

    // ---------------- Phase 2: context += w * x  via V_WMMA_F32_16X16X4_F32 --
    // D = A x B + C. A[16x4]: only row 0 carries weights (lane0: K0,K1 ;
    // lane16: K2,K3). B[4x16]: x rows t..t+3 over a 16-wide d-window, loaded
    // coalesced from row-major x (hot in L2 from phase 1). D row 0 (VGPR0,
    // lanes 0..15) accumulates the context slice.
    const bool hi   = lane >= 16;
    const int  n    = lane & 15;
    const bool row0 = (n == 0);

    for (int wi = wave; wi < DIM / 16; wi += WAVES) {
        const int dbase = wi * 16;
        v8f acc = {0.f, 0.f, 0.f, 0.f, 0.f, 0.f, 0.f, 0.f};
        for (int t = 0; t < TOKWG; t += 4) {
            float wlo = w_s[t + (hi ? 2 : 0)];
            float wh  = w_s[t + (hi ? 3 : 1)];
            v2f A;
            A.x = row0 ? wlo : 0.0f;
            A.y = row0 ? wh  : 0.0f;
            const float* bp = xb + (size_t)(tok0 + t + (hi ? 2 : 0)) * DIM + dbase + n;
            v2f B;
            B.x = bp[0];        // K = 0 (lanes 0-15) / K = 2 (lanes 16-31)
            B.y = bp[DIM];      // K = 1 (lanes 0-15) / K = 3 (lanes 16-31)
            acc = __builtin_amdgcn_wmma_f32_16x16x4_f32(
                false, A, false, B, (short)0, acc, false, false);
        }
        if (lane < 16) atomicAdd(&ctx_acc[b * DIM + dbase + lane], acc[0]);
    }
}

__global__ void normalize_out(const float* __restrict__ ctx_acc,
                              const float* __restrict__ l_acc,
                              float* __restrict__ out) {
    int i = blockIdx.x * blockDim.x + threadIdx.x;   // 0 .. BATCH*DIM-1
    if (i < BATCH * DIM) {
        int b = i / DIM;
        out[i] = ctx_acc[i] / l_acc[b];
    }
}

extern "C" void kernel_launch(void* const* d_in, const int* in_sizes, int n_in,
                              void* d_out, int out_size, void* d_ws, size_t ws_size,
                              hipStream_t stream) {
    const float* x  = (const float*)d_in[0];
    const float* Wa = (const float*)d_in[1];
    const float* ba = (const float*)d_in[2];
    const float* Wg = (const float*)d_in[3];
    const float* bg = (const float*)d_in[4];

    float* ws      = (float*)d_ws;
    float* ctx_acc = ws;                    // BATCH*DIM floats
    float* l_acc   = ws + BATCH * DIM;      // BATCH floats

    const int nz = BATCH * DIM + BATCH;
    zero_ws_kernel<<<(nz + 255) / 256, 256, 0, stream>>>(ws, nz);

    gated_attn_main<<<BATCH * CHUNKS, THREADS, 0, stream>>>(
        x, Wa, ba, Wg, bg, ctx_acc, l_acc);

    normalize_out<<<(BATCH * DIM + 255) / 256, 256, 0, stream>>>(
        ctx_acc, l_acc, (float*)d_out);
}